// Discriminator_712964571446
// MI455X (gfx1250) — compile-verified
//
#include <hip/hip_runtime.h>
#include <hip/hip_bf16.h>

typedef float v2f __attribute__((ext_vector_type(2)));
typedef float v8f __attribute__((ext_vector_type(8)));

#define NROWS 512
#define NCH   20
#define HID   128
#define NB    32
#define NC    16

// ---------------------------------------------------------------------------
// K1: fused relu(relu(x@W1+b1)@W2+b2) @ T  using V_WMMA_F32_16X16X4_F32.
// One block = 16 output rows, 4 waves. A/B fragments per CDNA5 ISA layout:
//   A 16x4 f32 : lanes 0-15 -> (M=lane, K=0..1), lanes 16-31 -> (M=lane-16, K=2..3)
//   B 4x16 f32 : lanes 0-15 -> (N=lane, K=0..1), lanes 16-31 -> (N=lane-16, K=2..3)
//   C/D 16x16  : lanes 0-15 -> (M=r,   N=lane),  lanes 16-31 -> (M=8+r, N=lane-16)
// ---------------------------------------------------------------------------
__global__ __launch_bounds__(128) void fused_mlp_wmma(
    const float* __restrict__ x,  const float* __restrict__ W1, const float* __restrict__ b1,
    const float* __restrict__ W2, const float* __restrict__ b2, const float* __restrict__ T,
    float* __restrict__ H, float* __restrict__ M3)
{
    __shared__ float xs[16 * NCH];        // 16x20 input tile
    __shared__ float w1s[NCH * HID];      // 20x128
    __shared__ float h1s[16 * HID];       // 16x128 intermediate
    __shared__ float w2s[HID * 32];       // 128x32 (N padded 20->32 with zeros)
    __shared__ float hs[16 * 32];         // 16x32 (K for GEMM3, padded)

    const int tid  = threadIdx.x;
    const int lane = tid & 31;
    const int wave = tid >> 5;
    const int r0   = blockIdx.x * 16;

    for (int idx = tid; idx < 16 * NCH; idx += 128) {
        int r = idx / NCH, c = idx % NCH;
        xs[idx] = x[(r0 + r) * NCH + c];
    }
    for (int idx = tid; idx < NCH * HID; idx += 128) w1s[idx] = W1[idx];
    for (int idx = tid; idx < HID * 32; idx += 128) {
        int k = idx >> 5, n = idx & 31;
        w2s[idx] = (n < NCH) ? W2[k * NCH + n] : 0.0f;
    }
    __syncthreads();

    const int m     = lane & 15;
    const int kk    = (lane >> 4) << 1;    // 0 or 2
    const int rbase = (lane >> 4) * 8;     // D rows 0..7 or 8..15

    // ---- GEMM1: h1 = relu(x @ W1 + b1), M=16 K=20 N=128 (8 n-tiles, 2/wave)
    for (int nt = wave * 2; nt < wave * 2 + 2; ++nt) {
        v8f acc = {};
        for (int k0 = 0; k0 < NCH; k0 += 4) {
            v2f a, b;
            a.x = xs[m * NCH + k0 + kk];
            a.y = xs[m * NCH + k0 + kk + 1];
            b.x = w1s[(k0 + kk) * HID + nt * 16 + m];
            b.y = w1s[(k0 + kk + 1) * HID + nt * 16 + m];
            acc = __builtin_amdgcn_wmma_f32_16x16x4_f32(false, a, false, b,
                                                        (short)0, acc, false, false);
        }
        int col = nt * 16 + m;
        float bias = b1[col];
        for (int r = 0; r < 8; ++r) {
            float v = acc[r] + bias;
            h1s[(rbase + r) * HID + col] = v > 0.0f ? v : 0.0f;
        }
    }
    __syncthreads();

    // ---- GEMM2: h = relu(h1 @ W2 + b2), M=16 K=128 N=32(pad), waves 0..1
    if (wave < 2) {
        const int nt = wave;
        v8f acc = {};
        for (int k0 = 0; k0 < HID; k0 += 4) {
            v2f a, b;
            a.x = h1s[m * HID + k0 + kk];
            a.y = h1s[m * HID + k0 + kk + 1];
            b.x = w2s[(k0 + kk) * 32 + nt * 16 + m];
            b.y = w2s[(k0 + kk + 1) * 32 + nt * 16 + m];
            acc = __builtin_amdgcn_wmma_f32_16x16x4_f32(false, a, false, b,
                                                        (short)0, acc, false, false);
        }
        int col = nt * 16 + m;
        float bias = (col < NCH) ? b2[col] : 0.0f;
        for (int r = 0; r < 8; ++r) {
            float v = acc[r] + bias;
            v = v > 0.0f ? v : 0.0f;
            hs[(rbase + r) * 32 + col] = v;                 // padded cols stay 0
            if (col < NCH) H[(r0 + rbase + r) * NCH + col] = v;
        }
    }
    __syncthreads();

    // ---- GEMM3: M = h @ T, M=16 K=20 N=512 (32 n-tiles, 8/wave); T via L2
    for (int nt = wave; nt < 32; nt += 4) {
        v8f acc = {};
        const int gn = nt * 16 + m;
        for (int k0 = 0; k0 < NCH; k0 += 4) {
            v2f a, b;
            a.x = hs[m * 32 + k0 + kk];
            a.y = hs[m * 32 + k0 + kk + 1];
            b.x = T[(k0 + kk) * NROWS + gn];
            b.y = T[(k0 + kk + 1) * NROWS + gn];
            acc = __builtin_amdgcn_wmma_f32_16x16x4_f32(false, a, false, b,
                                                        (short)0, acc, false, false);
        }
        for (int r = 0; r < 8; ++r)
            M3[(r0 + rbase + r) * NROWS + gn] = acc[r];
    }
}

// ---------------------------------------------------------------------------
// K2: o[i,b] = sum_j exp(-sum_c |M[i,b,c]-M[j,b,c]|). One block per row i.
// thread t: b = t&31 fixed, strides over j with step 8. Row i cached in LDS.
// ---------------------------------------------------------------------------
__global__ __launch_bounds__(256) void pairwise_o(const float* __restrict__ M3,
                                                  float* __restrict__ o)
{
    __shared__ float rowi[NROWS];
    __shared__ float red[256];
    const int i = blockIdx.x;
    const int t = threadIdx.x;
    for (int idx = t; idx < NROWS; idx += 256) rowi[idx] = M3[i * NROWS + idx];
    __syncthreads();

    const int b = t & 31;
    const int g = t >> 5;   // 0..7
    float s = 0.0f;
    for (int j = g; j < NROWS; j += 8) {
        const float* rj = &M3[j * NROWS + b * NC];
        float d = 0.0f;
#pragma unroll
        for (int c = 0; c < NC; ++c) d += fabsf(rowi[b * NC + c] - rj[c]);
        s += __expf(-d);
    }
    red[t] = s;
    __syncthreads();
    if (g == 0) {
        float tot = 0.0f;
#pragma unroll
        for (int gg = 0; gg < 8; ++gg) tot += red[gg * 32 + b];
        o[i * NB + b] = tot;
    }
}

// ---------------------------------------------------------------------------
// K3: mean of o, then out[i] = sigmoid([h_i, o_i - mean] @ W3 + b3).
// Single 512-thread block.
// ---------------------------------------------------------------------------
__global__ __launch_bounds__(512) void finalize(
    const float* __restrict__ H, const float* __restrict__ o,
    const float* __restrict__ W3, const float* __restrict__ b3,
    float* __restrict__ out)
{
    __shared__ float red[512];
    __shared__ float meansh;
    const int t = threadIdx.x;
    float s = 0.0f;
    for (int idx = t; idx < NROWS * NB; idx += 512) s += o[idx];
    red[t] = s;
    __syncthreads();
    for (int off = 256; off > 0; off >>= 1) {
        if (t < off) red[t] += red[t + off];
        __syncthreads();
    }
    if (t == 0) meansh = red[0] / (float)(NROWS * NB);
    __syncthreads();
    const float mean = meansh;

    float z = b3[0];
    for (int k = 0; k < NCH; ++k) z += H[t * NCH + k] * W3[k];
    for (int bb = 0; bb < NB; ++bb) z += (o[t * NB + bb] - mean) * W3[NCH + bb];
    out[t] = 1.0f / (1.0f + __expf(-z));
}

// ---------------------------------------------------------------------------
extern "C" void kernel_launch(void* const* d_in, const int* in_sizes, int n_in,
                              void* d_out, int out_size, void* d_ws, size_t ws_size,
                              hipStream_t stream) {
    const float* x  = (const float*)d_in[0];
    const float* W1 = (const float*)d_in[1];
    const float* b1 = (const float*)d_in[2];
    const float* W2 = (const float*)d_in[3];
    const float* b2 = (const float*)d_in[4];
    const float* T  = (const float*)d_in[5];
    const float* W3 = (const float*)d_in[6];
    const float* b3 = (const float*)d_in[7];
    float* out = (float*)d_out;

    char* ws = (char*)d_ws;
    float* H  = (float*)(ws);                       // 512*20*4  = 40 KB
    float* M3 = (float*)(ws + 65536);               // 512*512*4 = 1 MB
    float* o  = (float*)(ws + 65536 + 1048576);     // 512*32*4  = 64 KB

    fused_mlp_wmma<<<NROWS / 16, 128, 0, stream>>>(x, W1, b1, W2, b2, T, H, M3);
    pairwise_o<<<NROWS, 256, 0, stream>>>(M3, o);
    finalize<<<1, 512, 0, stream>>>(H, o, W3, b3, out);
}